// NCCWithGrad_48412871361081
// MI455X (gfx1250) — compile-verified
//
#include <hip/hip_runtime.h>

typedef __attribute__((ext_vector_type(2))) float v2f;
typedef __attribute__((ext_vector_type(8))) float v8f;

#define HH 512
#define WW 512
#define PLANES 32            // B*C = 16*2
#define ROWS_PER_STRIP 32
#define STRIPS 16            // 512 / 32
#define NCC_BLOCKS (PLANES * STRIPS)   // 512
#define NCC_THREADS 512
#define GRAD_BLOCKS 128
#define GRAD_THREADS 256
// ws float layout: [0..511] ncc partials, [512..639] dx partials, [640..767] dy partials
#define WS_DX 512
#define WS_DY 640

// ---------------------------------------------------------------------------
// NCC: per-column rolling vertical 9-sums in registers, horizontal 9-sum via LDS
// ---------------------------------------------------------------------------
__global__ __launch_bounds__(NCC_THREADS) void ncc_kernel(
    const float* __restrict__ I, const float* __restrict__ J,
    float* __restrict__ ws) {
  __shared__ float vsI[WW + 8], vsJ[WW + 8], vsII[WW + 8], vsJJ[WW + 8], vsIJ[WW + 8];
  __shared__ float red[NCC_THREADS];

  const int tid   = threadIdx.x;
  const int bid   = blockIdx.x;
  const int plane = bid >> 4;        // bid / STRIPS
  const int strip = bid & 15;
  const int y0    = strip * ROWS_PER_STRIP;
  const int x     = tid;             // column 0..511

  const float* __restrict__ Ip = I + plane * (HH * WW);
  const float* __restrict__ Jp = J + plane * (HH * WW);

  // zero-pad the horizontal halo once
  if (tid < 4) {
    vsI[tid] = 0.f;  vsJ[tid] = 0.f;  vsII[tid] = 0.f;  vsJJ[tid] = 0.f;  vsIJ[tid] = 0.f;
    vsI[WW + 4 + tid] = 0.f; vsJ[WW + 4 + tid] = 0.f; vsII[WW + 4 + tid] = 0.f;
    vsJJ[WW + 4 + tid] = 0.f; vsIJ[WW + 4 + tid] = 0.f;
  }

  float bi[9], bj[9];
#pragma unroll
  for (int k = 0; k < 9; ++k) { bi[k] = 0.f; bj[k] = 0.f; }
  float sI = 0.f, sJ = 0.f, sII = 0.f, sJJ = 0.f, sIJ = 0.f;
  float acc = 0.f;

  // warm-up: rows y0-4 .. y0+3 (rows < 0 are zero padding)
#pragma unroll
  for (int i = 0; i < 8; ++i) {
    const int r = y0 - 4 + i;
    float vi = 0.f, vj = 0.f;
    if (r >= 0) { vi = Ip[r * WW + x]; vj = Jp[r * WW + x]; }
    sI  += vi - bi[0];
    sJ  += vj - bj[0];
    sII += vi * vi - bi[0] * bi[0];
    sJJ += vj * vj - bj[0] * bj[0];
    sIJ += vi * vj - bi[0] * bj[0];
#pragma unroll
    for (int k = 0; k < 8; ++k) { bi[k] = bi[k + 1]; bj[k] = bj[k + 1]; }
    bi[8] = vi; bj[8] = vj;
  }

  for (int y = y0; y < y0 + ROWS_PER_STRIP; ++y) {
    const int r = y + 4;
    float vi = 0.f, vj = 0.f;
    if (r < HH) {
      vi = Ip[r * WW + x];
      vj = Jp[r * WW + x];
      if (r + 1 < HH) {  // CDNA5 global_prefetch_b8 for the next row
        __builtin_prefetch(&Ip[(r + 1) * WW + x], 0, 3);
        __builtin_prefetch(&Jp[(r + 1) * WW + x], 0, 3);
      }
    }
    // rolling vertical window update
    sI  += vi - bi[0];
    sJ  += vj - bj[0];
    sII += vi * vi - bi[0] * bi[0];
    sJJ += vj * vj - bj[0] * bj[0];
    sIJ += vi * vj - bi[0] * bj[0];
#pragma unroll
    for (int k = 0; k < 8; ++k) { bi[k] = bi[k + 1]; bj[k] = bj[k + 1]; }
    bi[8] = vi; bj[8] = vj;

    __syncthreads();  // previous iteration's LDS reads done (also covers pad init)
    vsI [4 + x] = sI;
    vsJ [4 + x] = sJ;
    vsII[4 + x] = sII;
    vsJJ[4 + x] = sJJ;
    vsIJ[4 + x] = sIJ;
    __syncthreads();

    float hI = 0.f, hJ = 0.f, hII = 0.f, hJJ = 0.f, hIJ = 0.f;
#pragma unroll
    for (int d = 0; d < 9; ++d) {
      hI  += vsI [x + d];
      hJ  += vsJ [x + d];
      hII += vsII[x + d];
      hJJ += vsJJ[x + d];
      hIJ += vsIJ[x + d];
    }
    // mirror the reference formula exactly
    const float u_I  = hI * (1.0f / 81.0f);
    const float u_J  = hJ * (1.0f / 81.0f);
    const float cross = hIJ - u_J * hI - u_I * hJ + u_I * u_J * 81.0f;
    const float Ivar  = hII - 2.0f * u_I * hI + u_I * u_I * 81.0f;
    const float Jvar  = hJJ - 2.0f * u_J * hJ + u_J * u_J * 81.0f;
    acc += cross * cross / (Ivar * Jvar + 1e-5f);
  }

  // deterministic block reduction
  __syncthreads();
  red[tid] = acc;
  __syncthreads();
  for (int s = NCC_THREADS / 2; s > 0; s >>= 1) {
    if (tid < s) red[tid] += red[tid + s];
    __syncthreads();
  }
  if (tid == 0) ws[bid] = red[0];
}

// ---------------------------------------------------------------------------
// Grad (L2 smoothness) partial sums over dvf [16,2,256,256]
// ---------------------------------------------------------------------------
__global__ __launch_bounds__(GRAD_THREADS) void grad_kernel(
    const float* __restrict__ V, float* __restrict__ ws) {
  __shared__ float rdx[GRAD_THREADS], rdy[GRAD_THREADS];
  const int N = 16 * 2 * 256 * 256;
  float adx = 0.f, ady = 0.f;
  for (int idx = blockIdx.x * blockDim.x + threadIdx.x; idx < N;
       idx += gridDim.x * blockDim.x) {
    const int w = idx & 255;
    const int h = (idx >> 8) & 255;
    const float v = V[idx];
    if (w < 255) { const float d = V[idx + 1]   - v; adx += d * d; }
    if (h < 255) { const float d = V[idx + 256] - v; ady += d * d; }
  }
  rdx[threadIdx.x] = adx;
  rdy[threadIdx.x] = ady;
  __syncthreads();
  for (int s = GRAD_THREADS / 2; s > 0; s >>= 1) {
    if (threadIdx.x < s) {
      rdx[threadIdx.x] += rdx[threadIdx.x + s];
      rdy[threadIdx.x] += rdy[threadIdx.x + s];
    }
    __syncthreads();
  }
  if (threadIdx.x == 0) {
    ws[WS_DX + blockIdx.x] = rdx[0];
    ws[WS_DY + blockIdx.x] = rdy[0];
  }
}

// ---------------------------------------------------------------------------
// Final reduction: chained V_WMMA_F32_16X16X4_F32 with B = all-ones.
// D[m][n] = sum_k A[m][k], so grand total = column-0 sum = s(lane0)+s(lane16)
// (uses only the documented C/D VGPR layout; A placement is irrelevant for a
// full sum, B=ones is layout-invariant).
// ---------------------------------------------------------------------------
__device__ inline float wave_wmma_sum(const float* __restrict__ p, int chunks) {
  v8f acc = {0.f, 0.f, 0.f, 0.f, 0.f, 0.f, 0.f, 0.f};
  const v2f b = {1.0f, 1.0f};
  const int lane = threadIdx.x & 31;
  for (int c = 0; c < chunks; ++c) {
    v2f a;
    a.x = p[c * 64 + lane];
    a.y = p[c * 64 + 32 + lane];
    acc = __builtin_amdgcn_wmma_f32_16x16x4_f32(
        /*neg_a=*/false, a, /*neg_b=*/false, b,
        /*c_mod=*/(short)0, acc, /*reuse_a=*/false, /*reuse_b=*/false);
  }
  const float s = acc[0] + acc[1] + acc[2] + acc[3] + acc[4] + acc[5] + acc[6] + acc[7];
  return __shfl(s, 0, 32) + __shfl(s, 16, 32);
}

__global__ __launch_bounds__(32) void finalize_kernel(
    const float* __restrict__ ws, float* __restrict__ out) {
  const float sumcc = wave_wmma_sum(ws,          8);  // 512 ncc partials
  const float sumdx = wave_wmma_sum(ws + WS_DX,  2);  // 128 dx partials
  const float sumdy = wave_wmma_sum(ws + WS_DY,  2);  // 128 dy partials
  if (threadIdx.x == 0) {
    const float ncc  = -sumcc * (1.0f / 4194304.0f);              // / (B*H*W), C-1 == 1
    const float grad = 0.01f * (sumdx * (1.0f / 2088960.0f) +
                                sumdy * (1.0f / 2088960.0f));     // 0.01 * 2 * (mx+my)/2
    out[0] = ncc + grad;
    out[1] = ncc;
    out[2] = grad;
  }
}

extern "C" void kernel_launch(void* const* d_in, const int* in_sizes, int n_in,
                              void* d_out, int out_size, void* d_ws, size_t ws_size,
                              hipStream_t stream) {
  (void)in_sizes; (void)n_in; (void)out_size; (void)ws_size;
  const float* y_true = (const float*)d_in[0];
  const float* y_pred = (const float*)d_in[1];
  const float* dvf    = (const float*)d_in[2];
  float* ws  = (float*)d_ws;
  float* out = (float*)d_out;

  ncc_kernel<<<NCC_BLOCKS, NCC_THREADS, 0, stream>>>(y_true, y_pred, ws);
  grad_kernel<<<GRAD_BLOCKS, GRAD_THREADS, 0, stream>>>(dvf, ws);
  finalize_kernel<<<1, 32, 0, stream>>>(ws, out);
}